// PatchExtractor_62294205661217
// MI455X (gfx1250) — compile-verified
//
#include <hip/hip_runtime.h>

typedef __attribute__((ext_vector_type(2))) float v2f;
typedef __attribute__((ext_vector_type(8))) float v8f;

#define BATCH 16
#define HH 512
#define WW 512
#define CC 3
#define OUTSZ 512

__device__ __forceinline__ int imin(int a, int b) { return a < b ? a : b; }
__device__ __forceinline__ int imax(int a, int b) { return a > b ? a : b; }
// Catmull-Rom style cubic kernel pieces (match reference exactly)
__device__ __forceinline__ float cub_k1(float d) { return (1.5f * d - 2.5f) * d * d + 1.0f; }
__device__ __forceinline__ float cub_k2(float d) { return ((-0.5f * d + 2.5f) * d - 4.0f) * d + 2.0f; }

// One wave32 per 16x16 output tile of one image.
// OUT_tile(16x16,c) = A(16xKr rowweights) * P_c(KrxKc patch) * B(Kcx16 colweights)
// Kr,Kc <= 12 (padded to 16), contraction via 4x V_WMMA_F32_16X16X4_F32 each.
__global__ __launch_bounds__(32) void patch_resize_wmma(
    const float* __restrict__ img, const int* __restrict__ coord,
    float* __restrict__ out)
{
    const int tc   = blockIdx.x;   // output col tile (16 cols)
    const int tr   = blockIdx.y;   // output row tile (16 rows)
    const int b    = blockIdx.z;   // image
    const int lane = threadIdx.x;  // 0..31
    const int m    = lane & 15;    // row (A/D "M") or col ("N") index
    const int half = lane >> 4;    // selects K pair in A/B frags, M+8 in D

    const int pw = coord[3 * b + 0];
    const int i1 = coord[3 * b + 1];
    const int i2 = coord[3 * b + 2];
    const float scl = (float)pw * (1.0f / (float)OUTSZ);

    __shared__ float Amat[256];    // [m][k]  row-weight band, 16x16
    __shared__ float Bmat[256];    // [k][n]  col-weight band, 16x16
    __shared__ float Pp[3][256];   // [k][j]  gathered patch window per channel
    __shared__ float Tt[256];      // [m][j]  intermediate A*P (reused per channel)

    // ---- zero weight matrices ----
#pragma unroll
    for (int i = 0; i < 8; ++i) {
        Amat[lane + 32 * i] = 0.0f;
        Bmat[lane + 32 * i] = 0.0f;
    }

    // ---- window bases (uniform per tile; every lane recomputes) ----
    const int ib_r = (int)floorf(((float)(tr * 16) + 0.5f) * scl - 0.5f);
    const int rb   = imin(imax(ib_r - 1, 0), pw - 1);   // patch-local first row
    const int ib_c = (int)floorf(((float)(tc * 16) + 0.5f) * scl - 0.5f);
    const int cb   = imin(imax(ib_c - 1, 0), pw - 1);   // patch-local first col

    __syncthreads();

    // ---- build banded weight matrices: lanes 0-15 -> rows, 16-31 -> cols ----
    {
        const int o    = (half == 0) ? (tr * 16 + m) : (tc * 16 + m);
        const float sr = ((float)o + 0.5f) * scl - 0.5f;
        const float f0 = floorf(sr);
        const float t  = sr - f0;
        const int  i0  = (int)f0;
        float w[4];
        w[0] = cub_k2(1.0f + t);
        w[1] = cub_k1(t);
        w[2] = cub_k1(1.0f - t);
        w[3] = cub_k2(2.0f - t);
        const int base = (half == 0) ? rb : cb;
#pragma unroll
        for (int f = 0; f < 4; ++f) {
            const int idx = imin(imax(i0 - 1 + f, 0), pw - 1);
            const int k   = idx - base;               // 0..15 (monotone => >= 0)
            if (half == 0) Amat[m * 16 + k] += w[f];  // lane owns row m
            else           Bmat[k * 16 + m] += w[f];  // lane owns col m
        }
    }
    __syncthreads();

    // ---- gather 16x16 input window (3 channels); weights are 0 on padding ----
    const float* imgb = img + (size_t)b * HH * WW * CC;
#pragma unroll
    for (int it = 0; it < 8; ++it) {
        const int p  = lane + 32 * it;
        const int k  = p >> 4;
        const int j  = p & 15;
        const int gr = imin(rb + i1 + k, HH - 1);     // clamp: padded taps have w=0
        const int gc = imin(cb + i2 + j, WW - 1);
        const float* px = imgb + ((size_t)gr * WW + gc) * CC;
        Pp[0][p] = px[0];
        Pp[1][p] = px[1];
        Pp[2][p] = px[2];
    }
    __syncthreads();

    // ---- A fragments (row weights) & B fragments (col weights), 4 K-chunks ----
    v2f afr[4], bcol[4];
#pragma unroll
    for (int kc = 0; kc < 4; ++kc) {
        afr[kc].x  = Amat[m * 16 + 4 * kc + 2 * half + 0];
        afr[kc].y  = Amat[m * 16 + 4 * kc + 2 * half + 1];
        bcol[kc].x = Bmat[(4 * kc + 2 * half + 0) * 16 + m];
        bcol[kc].y = Bmat[(4 * kc + 2 * half + 1) * 16 + m];
    }

    v8f dreg[3];
#pragma unroll
    for (int c = 0; c < 3; ++c) {
        // GEMM1: T = A(16xK) x P(Kx16), K=16 as 4 chunks of 4
        v8f acc = {};
#pragma unroll
        for (int kc = 0; kc < 4; ++kc) {
            v2f bfr;
            bfr.x = Pp[c][(4 * kc + 2 * half + 0) * 16 + m];
            bfr.y = Pp[c][(4 * kc + 2 * half + 1) * 16 + m];
            acc = __builtin_amdgcn_wmma_f32_16x16x4_f32(
                false, afr[kc], false, bfr, (short)0, acc, false, false);
        }
        __syncthreads();                      // Tt free from previous channel
        // D-layout -> LDS row-major: vgpr j holds M=j(+8*half), N=m
#pragma unroll
        for (int j = 0; j < 8; ++j)
            Tt[(j + 8 * half) * 16 + m] = acc[j];
        __syncthreads();

        // GEMM2: OUT = T(16xK) x B(Kx16)
        v8f acc2 = {};
#pragma unroll
        for (int kc = 0; kc < 4; ++kc) {
            v2f afr2;
            afr2.x = Tt[m * 16 + 4 * kc + 2 * half + 0];
            afr2.y = Tt[m * 16 + 4 * kc + 2 * half + 1];
            acc2 = __builtin_amdgcn_wmma_f32_16x16x4_f32(
                false, afr2, false, bcol[kc], (short)0, acc2, false, false);
        }
        dreg[c] = acc2;
    }

    // ---- store: lane holds N=m, M=j+8*half for vgpr j; 3 channels contiguous ----
    const int ocol = tc * 16 + m;
#pragma unroll
    for (int j = 0; j < 8; ++j) {
        const int orow = tr * 16 + j + 8 * half;
        float* po = out + ((size_t)(b * OUTSZ + orow) * OUTSZ + ocol) * CC;
        po[0] = dreg[0][j];
        po[1] = dreg[1][j];
        po[2] = dreg[2][j];
    }
}

extern "C" void kernel_launch(void* const* d_in, const int* in_sizes, int n_in,
                              void* d_out, int out_size, void* d_ws, size_t ws_size,
                              hipStream_t stream) {
    const float* img   = (const float*)d_in[0];
    const int*   coord = (const int*)d_in[1];
    float*       out   = (float*)d_out;
    dim3 grid(OUTSZ / 16, OUTSZ / 16, BATCH);  // 32 x 32 x 16 tiles
    patch_resize_wmma<<<grid, dim3(32, 1, 1), 0, stream>>>(img, coord, out);
}